// URGCNLayer_64854006169647
// MI455X (gfx1250) — compile-verified
//
#include <hip/hip_runtime.h>
#include <hip/hip_bf16.h>

typedef __attribute__((ext_vector_type(2))) float v2f;
typedef __attribute__((ext_vector_type(8))) float v8f;

#define DIM 128
#define ROWS_PER_BLOCK 80          // 5 M-tiles of 16 rows; 50000 % 80 == 0
#define MTILES 5

// ---------------------------------------------------------------------------
// K1: zero the scatter workspace (accum[N][128] + cnt[N])
// ---------------------------------------------------------------------------
__global__ void urgcn_zero_kernel(float* ws, long long n) {
    long long i = (long long)blockIdx.x * blockDim.x + threadIdx.x;
    if (i < n) ws[i] = 0.0f;
}

// ---------------------------------------------------------------------------
// K2: edge scatter. One wave32 per edge; each lane moves 4 floats (float4).
// accum[dst] += nodes[src] + rel_embed[rel];  cnt[dst] += 1.
// nodes (25.6MB) + accum (25.6MB) + rels (256KB) all sit in the 192MB L2,
// so the gather and the f32 atomics resolve in L2, not HBM.
// ---------------------------------------------------------------------------
__global__ void urgcn_edge_scatter(const int* __restrict__ edges,
                                   const float* __restrict__ nodes,
                                   const float* __restrict__ rels,
                                   float* __restrict__ accum,
                                   float* __restrict__ cnt,
                                   int n_edges) {
    int gid  = blockIdx.x * blockDim.x + threadIdx.x;
    int e    = gid >> 5;          // one wave (32 lanes) per edge
    int lane = gid & 31;
    if (e >= n_edges) return;

    int src = edges[3 * e + 0];
    int rel = edges[3 * e + 1];
    int dst = edges[3 * e + 2];

    const float4* sp = (const float4*)(nodes + (long long)src * DIM);
    const float4* rp = (const float4*)(rels  + (long long)rel * DIM);
    float4 s = sp[lane];
    float4 r = rp[lane];

    float* ap = accum + (long long)dst * DIM + lane * 4;
    atomicAdd(ap + 0, s.x + r.x);
    atomicAdd(ap + 1, s.y + r.y);
    atomicAdd(ap + 2, s.z + r.z);
    atomicAdd(ap + 3, s.w + r.w);
    if (lane == 0) atomicAdd(&cnt[dst], 1.0f);
}

// ---------------------------------------------------------------------------
// K3: fused node-level GEMM + residual, full f32 via V_WMMA_F32_16X16X4_F32.
//
// out = nodes + (mask * accum/cnt) @ Wn + (mask * nodes) @ Ws
//             + ((1-mask) * nodes) @ Wse
//
// One block = 80 output rows (5 M-tiles). The three masked A stripes
// (3 x 80 x 128 f32 = 120 KB) are staged once in LDS. Each of the 8 waves
// owns one 16-column N-tile and keeps 5 accumulator tiles in VGPRs, so every
// B fragment loaded from global (L2-hot) is reused by 5 wmma ops:
// per k-step = 2 global_load_b32 (B) + 5 ds_load_b64 (A) + 5 wmma.
//
// Fragment layouts per CDNA5 ISA 7.12.2:
//   A (16x4 f32, v2f): lanes 0-15 -> M=lane, K={0,1}; lanes 16-31 -> K={2,3}
//   B (4x16 f32, v2f): lanes 0-15 -> N=lane, K={0,1}; lanes 16-31 -> K={2,3}
//   C/D (16x16 f32, v8f): VGPR r -> row r (lanes 0-15) / row r+8 (lanes 16-31)
// ---------------------------------------------------------------------------
__global__ void __launch_bounds__(256)
urgcn_fused_gemm(const float* __restrict__ nodes,
                 const float* __restrict__ accum,
                 const float* __restrict__ cnt,
                 const float* __restrict__ wn,
                 const float* __restrict__ ws,
                 const float* __restrict__ wse,
                 float* __restrict__ out,
                 int n_nodes) {
    __shared__ float Abuf[3][ROWS_PER_BLOCK][DIM];   // 120 KB of 320 KB LDS

    const int m0   = blockIdx.x * ROWS_PER_BLOCK;
    const int tid  = threadIdx.x;
    const int wave = tid >> 5;
    const int lane = tid & 31;

    // Hint the (L2-hot) weight matrices toward the near caches.
    if (tid == 0) {
        __builtin_prefetch(wn,  0, 0);
        __builtin_prefetch(ws,  0, 0);
        __builtin_prefetch(wse, 0, 0);
    }

    // --- Stage masked A variants into LDS (10 rows per wave, float4 lanes) ---
    for (int r = wave; r < ROWS_PER_BLOCK; r += 8) {
        int row = m0 + r;
        if (row >= n_nodes) row = n_nodes - 1;   // clamp (stores are guarded)
        const int c4 = lane * 4;
        float c   = cnt[row];
        bool  has = c > 0.0f;
        float inv = has ? (1.0f / c) : 0.0f;     // agg mask folded into scale
        float4 nv = *(const float4*)(nodes + (long long)row * DIM + c4);
        float4 av = *(const float4*)(accum + (long long)row * DIM + c4);
        float4 a0 = make_float4(av.x * inv, av.y * inv, av.z * inv, av.w * inv);
        float4 a1 = has ? nv : make_float4(0.f, 0.f, 0.f, 0.f);
        float4 a2 = has ? make_float4(0.f, 0.f, 0.f, 0.f) : nv;
        *(float4*)&Abuf[0][r][c4] = a0;
        *(float4*)&Abuf[1][r][c4] = a1;
        *(float4*)&Abuf[2][r][c4] = a2;
    }
    __syncthreads();

    // --- Chained WMMA: K = 3 * 128, five f32 accumulator tiles per wave ---
    const int n0    = wave * 16;          // this wave's 16 output columns
    const int lrow  = lane & 15;          // M row / N col within fragment
    const int khalf = lane >> 4;          // K sub-pair selector (0 or 1)
    const int col   = n0 + lrow;

    v8f acc0 = {}, acc1 = {}, acc2 = {}, acc3 = {}, acc4 = {};
    const float* Wmat[3] = { wn, ws, wse };

    #pragma unroll
    for (int g = 0; g < 3; ++g) {
        const float* __restrict__ W = Wmat[g];
        #pragma unroll 2
        for (int kk = 0; kk < 32; ++kk) {
            const int k0 = kk * 4 + khalf * 2;
            v2f b;
            b.x = W[(long long)k0 * DIM + col];        // W[k0][col]
            b.y = W[(long long)(k0 + 1) * DIM + col];  // W[k0+1][col]

            v2f a;
            a.x = Abuf[g][lrow +  0][k0];  a.y = Abuf[g][lrow +  0][k0 + 1];
            acc0 = __builtin_amdgcn_wmma_f32_16x16x4_f32(false, a, false, b,
                                                         (short)0, acc0, false, false);
            a.x = Abuf[g][lrow + 16][k0];  a.y = Abuf[g][lrow + 16][k0 + 1];
            acc1 = __builtin_amdgcn_wmma_f32_16x16x4_f32(false, a, false, b,
                                                         (short)0, acc1, false, false);
            a.x = Abuf[g][lrow + 32][k0];  a.y = Abuf[g][lrow + 32][k0 + 1];
            acc2 = __builtin_amdgcn_wmma_f32_16x16x4_f32(false, a, false, b,
                                                         (short)0, acc2, false, false);
            a.x = Abuf[g][lrow + 48][k0];  a.y = Abuf[g][lrow + 48][k0 + 1];
            acc3 = __builtin_amdgcn_wmma_f32_16x16x4_f32(false, a, false, b,
                                                         (short)0, acc3, false, false);
            a.x = Abuf[g][lrow + 64][k0];  a.y = Abuf[g][lrow + 64][k0 + 1];
            acc4 = __builtin_amdgcn_wmma_f32_16x16x4_f32(false, a, false, b,
                                                         (short)0, acc4, false, false);
        }
    }

    // --- Epilogue: residual add + guarded store ---
    #pragma unroll
    for (int t = 0; t < MTILES; ++t) {
        const v8f* accp = (t == 0) ? &acc0 : (t == 1) ? &acc1 :
                          (t == 2) ? &acc2 : (t == 3) ? &acc3 : &acc4;
        #pragma unroll
        for (int r = 0; r < 8; ++r) {
            const int row = m0 + t * 16 + r + khalf * 8;
            if (row < n_nodes) {
                out[(long long)row * DIM + col] =
                    nodes[(long long)row * DIM + col] + (*accp)[r];
            }
        }
    }
}

// ---------------------------------------------------------------------------
extern "C" void kernel_launch(void* const* d_in, const int* in_sizes, int n_in,
                              void* d_out, int out_size, void* d_ws, size_t ws_size,
                              hipStream_t stream) {
    const float* nodes = (const float*)d_in[0];   // [N, 128]
    const float* rels  = (const float*)d_in[1];   // [R, 128]
    const int*   edges = (const int*)d_in[2];     // [E, 3]
    const float* wn    = (const float*)d_in[3];   // [128, 128]
    const float* ws_m  = (const float*)d_in[4];   // [128, 128]
    const float* wse   = (const float*)d_in[5];   // [128, 128]
    float*       out   = (float*)d_out;

    const int n_nodes = in_sizes[0] / DIM;
    const int n_edges = in_sizes[2] / 3;

    float* accum = (float*)d_ws;                         // [N, 128]
    float* cntf  = accum + (long long)n_nodes * DIM;     // [N]

    // K1: zero accum + cnt
    long long nz = (long long)n_nodes * DIM + n_nodes;
    {
        int blocks = (int)((nz + 255) / 256);
        urgcn_zero_kernel<<<blocks, 256, 0, stream>>>((float*)d_ws, nz);
    }
    // K2: edge scatter (one wave per edge; 8 edges per 256-thread block)
    {
        int blocks = (n_edges + 7) / 8;
        urgcn_edge_scatter<<<blocks, 256, 0, stream>>>(edges, nodes, rels,
                                                       accum, cntf, n_edges);
    }
    // K3: fused GEMM + residual (one block per 80 rows; 50000/80 = 625)
    {
        int blocks = (n_nodes + ROWS_PER_BLOCK - 1) / ROWS_PER_BLOCK;
        urgcn_fused_gemm<<<blocks, 256, 0, stream>>>(nodes, accum, cntf,
                                                     wn, ws_m, wse, out,
                                                     n_nodes);
    }
}